// WeightedPnPHead_42219528520115
// MI455X (gfx1250) — compile-verified
//
#include <hip/hip_runtime.h>
#include <hip/hip_bf16.h>
#include <math.h>

// MI455X / gfx1250, wave32. fp32 WMMA: V_WMMA_F32_16X16X4_F32
typedef float v2f __attribute__((ext_vector_type(2)));
typedef float v8f __attribute__((ext_vector_type(8)));

#define NPTS 512           // points per batch
#define WAVES_PER_BLOCK 8  // 256 threads = 8 wave32

// ---------------------------------------------------------------------------
// Pass 1: streaming Gram accumulation (memory-roofline pass, ~50 MB total).
// Per point: r = w*[x,y,z,1, u*x,u*y,u*z,u, x,y,z,1, v*x,v*y,v*z,v]  (16 floats)
// D = sum_r r r^T  via v_wmma_f32_16x16x4_f32, same fragment as A and B
// (symmetric outer product; A 16x4 and B 4x16 layouts mirror).
// Each wave owns one batch and a private LDS region: no __syncthreads needed;
// per-wave LDS ordering + compiler may-alias waits (s_wait_dscnt) suffice.
// Also tracks per-batch argmax(weight) point (x,y,z) for the sign fix
// (branchless selects, no exec divergence in the hot loop).
// ---------------------------------------------------------------------------
__global__ __launch_bounds__(256)
void pnp_gram_wmma(const float* __restrict__ src,
                   const float* __restrict__ corres,
                   const float* __restrict__ wts,
                   float* __restrict__ Dws,      // [nB][256]
                   float* __restrict__ bestws)   // [nB][4] = x,y,z,(unused)
{
  const int wave = threadIdx.x >> 5;
  const int lane = threadIdx.x & 31;
  const int b    = blockIdx.x * WAVES_PER_BLOCK + wave;

  __shared__ float lds[WAVES_PER_BLOCK][32][16];   // 16 KB: 32 points x 16 comps / wave

  const float* sb = src    + (size_t)b * NPTS * 3;
  const float* cb = corres + (size_t)b * NPTS * 2;
  const float* wb = wts    + (size_t)b * NPTS;

  v8f acc = {0.f,0.f,0.f,0.f,0.f,0.f,0.f,0.f};

  float bw = -1e30f; int bi = 1 << 30;
  float bx = 0.f, by = 0.f, bz = 0.f;

  const int hi = lane >> 4;     // 0 for lanes 0-15, 1 for 16-31
  const int lo = lane & 15;

  for (int t = 0; t < NPTS / 32; ++t) {
    const int n = t * 32 + lane;
    const float x = sb[n*3+0], y = sb[n*3+1], z = sb[n*3+2];
    const float u = cb[n*2+0], v = cb[n*2+1];
    const float w = wb[n];
    if (t + 1 < NPTS / 32) {  // global_prefetch_b8 for next tile
      __builtin_prefetch(sb + (n + 32) * 3, 0, 1);
      __builtin_prefetch(cb + (n + 32) * 2, 0, 1);
    }

    // branchless first-max argmax update (v_cndmask, no exec divergence)
    {
      const bool better = (w > bw) | ((w == bw) & (n < bi));
      bw = better ? w : bw;
      bi = better ? n : bi;
      bx = better ? x : bx;
      by = better ? y : by;
      bz = better ? z : bz;
    }

    float* p = &lds[wave][lane][0];
    const float wu = w * u, wv = w * v;
    p[0]  = w*x;  p[1]  = w*y;  p[2]  = w*z;  p[3]  = w;
    p[4]  = wu*x; p[5]  = wu*y; p[6]  = wu*z; p[7]  = wu;
    p[8]  = w*x;  p[9]  = w*y;  p[10] = w*z;  p[11] = w;
    p[12] = wv*x; p[13] = wv*y; p[14] = wv*z; p[15] = wv;
    // no barrier: same-wave DS ops are in-order; compiler emits s_wait_dscnt

    // 8 groups of 4 points -> 8 symmetric-outer-product WMMAs
#pragma unroll
    for (int g = 0; g < 8; ++g) {
      v2f a;
      a[0] = lds[wave][4*g + 2*hi + 0][lo];
      a[1] = lds[wave][4*g + 2*hi + 1][lo];
      acc = __builtin_amdgcn_wmma_f32_16x16x4_f32(
          false, a, false, a, (short)0, acc, false, false);
    }
  }

  // D layout in VGPRs: acc[r] = D[row = r + 8*hi][col = lo]
#pragma unroll
  for (int r = 0; r < 8; ++r)
    Dws[(size_t)b * 256 + (size_t)(r + 8*hi) * 16 + lo] = acc[r];

  // wave argmax reduction (first-max tie-break, matches jnp.argmax)
#pragma unroll
  for (int off = 16; off >= 1; off >>= 1) {
    float ow = __shfl_xor(bw, off, 32);
    int   oi = __shfl_xor(bi, off, 32);
    float ox = __shfl_xor(bx, off, 32);
    float oy = __shfl_xor(by, off, 32);
    float oz = __shfl_xor(bz, off, 32);
    const bool take = (ow > bw) | ((ow == bw) & (oi < bi));
    bw = take ? ow : bw; bi = take ? oi : bi;
    bx = take ? ox : bx; by = take ? oy : by; bz = take ? oz : bz;
  }
  if (lane == 0) {
    bestws[b*4+0] = bx; bestws[b*4+1] = by; bestws[b*4+2] = bz; bestws[b*4+3] = 0.f;
  }
}

// assemble M(i,j) (12x12 normal matrix) from the 16x16 Gram D
__device__ __forceinline__ float Mij(const float* __restrict__ Dp, int i, int j) {
  if (i > j) { int t = i; i = j; j = t; }
  const int bi = i >> 2, bj = j >> 2, li = i & 3, lj = j & 3;
  if (bi == 0 && bj == 0) return  Dp[li*16 + lj];            // H
  if (bi == 0 && bj == 1) return  0.f;
  if (bi == 0 && bj == 2) return -Dp[li*16 + 4 + lj];        // -Hu
  if (bi == 1 && bj == 1) return  Dp[li*16 + lj];            // H
  if (bi == 1 && bj == 2) return -Dp[(8+li)*16 + 12 + lj];   // -Hv
  return Dp[(4+li)*16 + 4+lj] + Dp[(12+li)*16 + 12+lj];      // Hu2+Hv2
}

// ---------------------------------------------------------------------------
// Pass 2: per batch (one wave), smallest eigenvector of M via WMMA power
// iteration on P = s*I - M (s = Gershgorin bound => P PSD, dominant eigvec
// of P = nullspace direction of M), then closed-form 3x3 singular values,
// beta, sign fix, emit R (b*9) and t (nB*9 + b*3). Eigenvector sign
// ambiguity cancels through the z>=0 sign logic, matching the reference.
// ---------------------------------------------------------------------------
__global__ __launch_bounds__(256)
void pnp_solve_wmma(const float* __restrict__ Dws,
                    const float* __restrict__ bestws,
                    float* __restrict__ out, int nB)
{
  const int wave = threadIdx.x >> 5;
  const int lane = threadIdx.x & 31;
  const int b    = blockIdx.x * WAVES_PER_BLOCK + wave;
  const int hi   = lane >> 4, lo = lane & 15;

  __shared__ float Dm[WAVES_PER_BLOCK][256];   // 8 KB, per-wave private region
#pragma unroll
  for (int r = 0; r < 8; ++r)
    Dm[wave][(r + 8*hi)*16 + lo] = Dws[(size_t)b * 256 + (size_t)(r + 8*hi)*16 + lo];
  // no barrier: per-wave private LDS region, same-wave DS ordering suffices
  const float* Dp = &Dm[wave][0];

  // Gershgorin upper bound on lambda_max(M)
  float rs = 0.f;
  if (lane < 12) {
#pragma unroll
    for (int j = 0; j < 12; ++j) rs += fabsf(Mij(Dp, lane, j));
  }
#pragma unroll
  for (int off = 16; off >= 1; off >>= 1) rs = fmaxf(rs, __shfl_xor(rs, off, 32));
  const float s = rs;

  // loop-invariant A fragments of P = s*I - M (16x16, zero-padded rows/cols 12..15)
  v2f pa[4];
  const int row = lo;
#pragma unroll
  for (int c = 0; c < 4; ++c) {
    const int c0 = 4*c + 2*hi, c1 = c0 + 1;
    float p0 = ((row == c0) ? s : 0.f) - ((row < 12 && c0 < 12) ? Mij(Dp, row, c0) : 0.f);
    float p1 = ((row == c1) ? s : 0.f) - ((row < 12 && c1 < 12) ? Mij(Dp, row, c1) : 0.f);
    pa[c][0] = p0; pa[c][1] = p1;
  }

  float xs[16];
#pragma unroll
  for (int j = 0; j < 16; ++j) xs[j] = (j < 12) ? 0.28867513f : 0.f;  // unit start

  for (int it = 0; it < 64; ++it) {
    // B fragments: vector broadcast into all 16 columns
    v2f vb[4];
#pragma unroll
    for (int c = 0; c < 4; ++c) {
      vb[c][0] = hi ? xs[4*c + 2] : xs[4*c + 0];
      vb[c][1] = hi ? xs[4*c + 3] : xs[4*c + 1];
    }
    v8f y = {0.f,0.f,0.f,0.f,0.f,0.f,0.f,0.f};
#pragma unroll
    for (int c = 0; c < 4; ++c)
      y = __builtin_amdgcn_wmma_f32_16x16x4_f32(
          false, pa[c], false, vb[c], (short)0, y, false, false);

    // normalize: lane holds rows (r+8*hi) of its column; columns identical
    float nn = 0.f;
#pragma unroll
    for (int r = 0; r < 8; ++r) nn += y[r] * y[r];
    nn += __shfl_xor(nn, 16, 32);
    const float sc = rsqrtf(fmaxf(nn, 1e-30f));
    float yv[8];
#pragma unroll
    for (int r = 0; r < 8; ++r) yv[r] = y[r] * sc;

    // broadcast normalized vector to all lanes: x[j<8] from lane0, x[j>=8] from lane16
#pragma unroll
    for (int j = 0; j < 8; ++j) xs[j]     = __shfl(yv[j], 0, 32);
#pragma unroll
    for (int j = 0; j < 8; ++j) xs[8 + j] = __shfl(yv[j], 16, 32);
  }

  if (lane == 0) {
    // tilde_f = xs[0..11], F = reshape(3,4); F33 cols, t-col
    const float* v = xs;
    float f[3][3] = {{v[0],v[1],v[2]},{v[4],v[5],v[6]},{v[8],v[9],v[10]}};
    // E = F33^T F33
    float E00=0,E01=0,E02=0,E11=0,E12=0,E22=0;
#pragma unroll
    for (int i = 0; i < 3; ++i) {
      E00 += f[i][0]*f[i][0]; E01 += f[i][0]*f[i][1]; E02 += f[i][0]*f[i][2];
      E11 += f[i][1]*f[i][1]; E12 += f[i][1]*f[i][2]; E22 += f[i][2]*f[i][2];
    }
    // closed-form eigenvalues of symmetric 3x3 (trig method); s_i = sqrt(l_i)
    const float q  = (E00 + E11 + E22) * (1.f/3.f);
    const float pp1 = E01*E01 + E02*E02 + E12*E12;
    const float pp2 = (E00-q)*(E00-q) + (E11-q)*(E11-q) + (E22-q)*(E22-q) + 2.f*pp1;
    float l1, l2, l3;
    if (pp2 < 1e-24f) { l1 = l2 = l3 = q; }
    else {
      const float p  = sqrtf(pp2 * (1.f/6.f));
      const float ip = 1.f / p;
      const float b00=(E00-q)*ip, b01=E01*ip, b02=E02*ip;
      const float b11=(E11-q)*ip, b12=E12*ip, b22=(E22-q)*ip;
      float det = b00*(b11*b22 - b12*b12) - b01*(b01*b22 - b12*b02)
                + b02*(b01*b12 - b11*b02);
      float r2 = fminf(1.f, fmaxf(-1.f, det * 0.5f));
      const float phi = acosf(r2) * (1.f/3.f);
      l1 = q + 2.f*p*cosf(phi);
      l3 = q + 2.f*p*cosf(phi + 2.0943951023931953f);  // +2pi/3
      l2 = 3.f*q - l1 - l3;
    }
    const float ssum = sqrtf(fmaxf(l1,0.f)) + sqrtf(fmaxf(l2,0.f)) + sqrtf(fmaxf(l3,0.f));
    float beta = 1.f / (ssum * (1.f/3.f) + 1e-10f);

    // sign fix from argmax-weight point (homo = [x,y,z,1]), F row 2 = v[8..11]
    const float bx = bestws[b*4+0], by = bestws[b*4+1], bz = bestws[b*4+2];
    const float zz = beta * (bx*v[8] + by*v[9] + bz*v[10] + v[11]);
    beta = (zz >= 0.f) ? beta : -beta;

    float* Rout = out + (size_t)b * 9;
#pragma unroll
    for (int i = 0; i < 3; ++i)
#pragma unroll
      for (int j = 0; j < 3; ++j) Rout[i*3 + j] = beta * v[4*i + j];
    float* Tout = out + (size_t)nB * 9 + (size_t)b * 3;
#pragma unroll
    for (int i = 0; i < 3; ++i) Tout[i] = beta * v[4*i + 3];
  }
}

// ---------------------------------------------------------------------------
extern "C" void kernel_launch(void* const* d_in, const int* in_sizes, int n_in,
                              void* d_out, int out_size, void* d_ws, size_t ws_size,
                              hipStream_t stream) {
  const float* src    = (const float*)d_in[0];   // (B, N, 3) f32
  const float* corres = (const float*)d_in[1];   // (B, N, 2) f32
  const float* wts    = (const float*)d_in[2];   // (B, N)    f32
  float*       out    = (float*)d_out;           // R (B*9) then t (B*3)

  const int nB = in_sizes[2] / NPTS;             // 4096

  float* Dws    = (float*)d_ws;                  // nB * 256 floats (4 MB)
  float* bestws = Dws + (size_t)nB * 256;        // nB * 4 floats

  const int blocks = nB / WAVES_PER_BLOCK;       // 512
  pnp_gram_wmma<<<blocks, WAVES_PER_BLOCK * 32, 0, stream>>>(src, corres, wts, Dws, bestws);
  pnp_solve_wmma<<<blocks, WAVES_PER_BLOCK * 32, 0, stream>>>(Dws, bestws, out, nB);
}